// ConsistentMatcher_48301202211264
// MI455X (gfx1250) — compile-verified
//
#include <hip/hip_runtime.h>
#include <hip/hip_bf16.h>

typedef __attribute__((ext_vector_type(2))) float v2f;
typedef __attribute__((ext_vector_type(8))) float v8f;

#define N_   6144
#define M_   6144
#define D_   128
#define TILE 128
#define CT   (M_ / TILE)   // 48 column tiles
#define RT   (N_ / TILE)   // 48 row tiles
#define LDS_STRIDE 129     // pad to avoid LDS bank conflicts (64 banks)

// ---------------------------------------------------------------------------
// Squared norms of each descriptor row (reference computes them explicitly).
// ---------------------------------------------------------------------------
__global__ void sqnorm_kernel(const float* __restrict__ d1,
                              const float* __restrict__ d2,
                              float* __restrict__ sq1,
                              float* __restrict__ sq2) {
  int i = blockIdx.x * blockDim.x + threadIdx.x;
  const float* src;
  float* dst;
  int row;
  if (i < N_) {
    src = d1; dst = sq1; row = i;
  } else if (i < N_ + M_) {
    src = d2; dst = sq2; row = i - N_;
  } else {
    return;
  }
  const float4* p = (const float4*)(src + (size_t)row * D_);
  float s = 0.f;
#pragma unroll
  for (int j = 0; j < D_ / 4; ++j) {
    float4 v = p[j];
    s += v.x * v.x + v.y * v.y + v.z * v.z + v.w * v.w;
  }
  dst[row] = s;
}

// ---------------------------------------------------------------------------
// Core: compute one 128x128 affinity tile with f32 WMMA and dump to LDS.
//   affinity = -invT * sqrt(max(sq1[m]+sq2[n]-2*inner, 0) + 1e-12)
// Block = 256 threads = 8 wave32s in a 4x2 grid; wave region = 32 rows x 64
// cols = 2x4 subtiles of 16x16. K loop steps by 4 (V_WMMA_F32_16X16X4_F32).
// A fragment (16x4, MxK): lane l holds row (l%16), K = {k+2*(l/16), +1}.
// B fragment (4x16, KxN): lane l holds col (l%16), K = {k+2*(l/16), +1}
//   -> both are contiguous float2 loads from the row-major desc arrays.
// ---------------------------------------------------------------------------
__device__ __forceinline__ void compute_affinity_tile(
    const float* __restrict__ d1, const float* __restrict__ d2,
    const float* __restrict__ sq1, const float* __restrict__ sq2,
    float invT, int rowBase, int colBase, float* lds) {
  const int tid  = threadIdx.x;
  const int lane = tid & 31;
  const int w    = tid >> 5;   // 0..7
  const int wr   = w >> 1;     // 0..3  (32-row strip)
  const int wc   = w & 1;      // 0..1  (64-col strip)
  const int half = lane >> 4;  // 0/1
  const int l16  = lane & 15;

  v8f acc[2][4];
  const v8f zero = {0.f, 0.f, 0.f, 0.f, 0.f, 0.f, 0.f, 0.f};
#pragma unroll
  for (int sr = 0; sr < 2; ++sr)
#pragma unroll
    for (int sc = 0; sc < 4; ++sc) acc[sr][sc] = zero;

  const float* aP[2];
  const float* bP[4];
#pragma unroll
  for (int sr = 0; sr < 2; ++sr)
    aP[sr] = d1 + (size_t)(rowBase + wr * 32 + sr * 16 + l16) * D_ + 2 * half;
#pragma unroll
  for (int sc = 0; sc < 4; ++sc)
    bP[sc] = d2 + (size_t)(colBase + wc * 64 + sc * 16 + l16) * D_ + 2 * half;

#pragma unroll 4
  for (int k = 0; k < D_; k += 4) {
    v2f aF[2], bF[4];
#pragma unroll
    for (int sr = 0; sr < 2; ++sr) aF[sr] = *(const v2f*)(aP[sr] + k);
#pragma unroll
    for (int sc = 0; sc < 4; ++sc) bF[sc] = *(const v2f*)(bP[sc] + k);
#pragma unroll
    for (int sr = 0; sr < 2; ++sr)
#pragma unroll
      for (int sc = 0; sc < 4; ++sc)
        acc[sr][sc] = __builtin_amdgcn_wmma_f32_16x16x4_f32(
            false, aF[sr], false, bF[sc], (short)0, acc[sr][sc], false, false);
  }

  // Epilogue: distance -> affinity, dump to LDS.
  // C layout: VGPR g, lanes 0-15 -> M = g, lanes 16-31 -> M = g+8; N = lane%16.
#pragma unroll
  for (int sr = 0; sr < 2; ++sr)
#pragma unroll
    for (int sc = 0; sc < 4; ++sc) {
      int n_loc = wc * 64 + sc * 16 + l16;
      float s2  = sq2[colBase + n_loc];
#pragma unroll
      for (int g = 0; g < 8; ++g) {
        int   m_loc = wr * 32 + sr * 16 + half * 8 + g;
        float inner = acc[sr][sc][g];
        float dd    = sq1[rowBase + m_loc] + s2 - 2.0f * inner;
        dd          = fmaxf(dd, 0.0f) + 1e-12f;
        lds[m_loc * LDS_STRIDE + n_loc] = -invT * sqrtf(dd);
      }
    }
}

// ---------------------------------------------------------------------------
// Pass 1: per-tile row/col softmax partials (max, sum exp, argmax).
// Partials laid out [tile][element] so writes and the reduce reads coalesce.
// ---------------------------------------------------------------------------
__global__ void __launch_bounds__(256) gemm_stats_kernel(
    const float* __restrict__ d1, const float* __restrict__ d2,
    const float* __restrict__ sq1, const float* __restrict__ sq2,
    const float* __restrict__ invTp,
    float* __restrict__ rowMaxPart, float* __restrict__ rowSumPart,
    int* __restrict__ rowArgPart, float* __restrict__ colMaxPart,
    float* __restrict__ colSumPart, int* __restrict__ colArgPart) {
  __shared__ float lds[TILE * LDS_STRIDE];
  const int bx = blockIdx.x, by = blockIdx.y;
  const int rowBase = by * TILE, colBase = bx * TILE;
  compute_affinity_tile(d1, d2, sq1, sq2, invTp[0], rowBase, colBase, lds);
  __syncthreads();

  const int tid = threadIdx.x;
  if (tid < TILE) {
    // Threads 0..127: one row each (conflict-free: stride 129).
    const int r = tid;
    const float* p = &lds[r * LDS_STRIDE];
    float mx = p[0];
    int   arg = 0;
    for (int j = 1; j < TILE; ++j) {
      float v = p[j];
      if (v > mx) { mx = v; arg = j; }
    }
    float s = 0.f;
    for (int j = 0; j < TILE; ++j) s += expf(p[j] - mx);
    size_t o = (size_t)bx * N_ + rowBase + r;
    rowMaxPart[o] = mx;
    rowSumPart[o] = s;
    rowArgPart[o] = colBase + arg;
  } else {
    // Threads 128..255: one column each.
    const int c = tid - TILE;
    float mx = lds[c];
    int   arg = 0;
    for (int j = 1; j < TILE; ++j) {
      float v = lds[j * LDS_STRIDE + c];
      if (v > mx) { mx = v; arg = j; }
    }
    float s = 0.f;
    for (int j = 0; j < TILE; ++j) s += expf(lds[j * LDS_STRIDE + c] - mx);
    size_t o = (size_t)by * M_ + colBase + c;
    colMaxPart[o] = mx;
    colSumPart[o] = s;
    colArgPart[o] = rowBase + arg;
  }
}

// ---------------------------------------------------------------------------
// Combine per-tile partials -> off = max + log(sum exp) and global argmax.
// Tiles scanned in order with strict > keeps first-occurrence argmax semantics.
// ---------------------------------------------------------------------------
__global__ void reduce_stats_kernel(const float* __restrict__ maxPart,
                                    const float* __restrict__ sumPart,
                                    const int* __restrict__ argPart, int n,
                                    int parts, float* __restrict__ offOut,
                                    int* __restrict__ argOut) {
  int i = blockIdx.x * blockDim.x + threadIdx.x;
  if (i >= n) return;
  float mx = -__builtin_inff();
  int   arg = 0;
  for (int t = 0; t < parts; ++t) {
    float v = maxPart[(size_t)t * n + i];
    if (v > mx) { mx = v; arg = argPart[(size_t)t * n + i]; }
  }
  float s = 0.f;
  for (int t = 0; t < parts; ++t)
    s += sumPart[(size_t)t * n + i] * expf(maxPart[(size_t)t * n + i] - mx);
  offOut[i] = mx + logf(s);
  argOut[i] = arg;
}

// ---------------------------------------------------------------------------
// Pass 2: recompute affinity tile (bit-identical), emit
//   dense_logp = 2a - rowOff[m] - colOff[n],  dense_p = exp(dense_logp)
// as coalesced float4 stores.
// ---------------------------------------------------------------------------
__global__ void __launch_bounds__(256) gemm_write_kernel(
    const float* __restrict__ d1, const float* __restrict__ d2,
    const float* __restrict__ sq1, const float* __restrict__ sq2,
    const float* __restrict__ invTp, const float* __restrict__ rowOff,
    const float* __restrict__ colOff, float* __restrict__ dense_logp,
    float* __restrict__ dense_p) {
  __shared__ float lds[TILE * LDS_STRIDE];
  const int bx = blockIdx.x, by = blockIdx.y;
  const int rowBase = by * TILE, colBase = bx * TILE;
  compute_affinity_tile(d1, d2, sq1, sq2, invTp[0], rowBase, colBase, lds);
  __syncthreads();

  const int tid = threadIdx.x;
#pragma unroll
  for (int iter = 0; iter < 16; ++iter) {
    int f  = iter * 256 + tid;   // float4 chunk id, 4096 per tile
    int m  = f >> 5;             // 32 float4 per 128-wide row
    int n4 = f & 31;
    const float* p = &lds[m * LDS_STRIDE + n4 * 4];
    float ro  = rowOff[rowBase + m];
    int  ncol = colBase + n4 * 4;
    float4 lp, pp;
    lp.x = 2.f * p[0] - ro - colOff[ncol + 0];
    lp.y = 2.f * p[1] - ro - colOff[ncol + 1];
    lp.z = 2.f * p[2] - ro - colOff[ncol + 2];
    lp.w = 2.f * p[3] - ro - colOff[ncol + 3];
    pp.x = expf(lp.x); pp.y = expf(lp.y); pp.z = expf(lp.z); pp.w = expf(lp.w);
    size_t o = (size_t)(rowBase + m) * M_ + ncol;
    *(float4*)(dense_logp + o) = lp;
    *(float4*)(dense_p + o)    = pp;
  }
}

// ---------------------------------------------------------------------------
// matches [2,N] (as float, per output dtype) + cycle-consistency mask [N].
// ---------------------------------------------------------------------------
__global__ void matches_kernel(const int* __restrict__ rowArg,
                               const int* __restrict__ colArg,
                               float* __restrict__ out) {
  int i = blockIdx.x * blockDim.x + threadIdx.x;
  if (i >= N_) return;
  int mi = rowArg[i];
  out[0 * N_ + i] = (float)i;                           // matches row 0: idx
  out[1 * N_ + i] = (float)mi;                          // matches row 1: maxes_I
  out[2 * N_ + i] = (colArg[mi] == i) ? 1.0f : 0.0f;    // cycle_consistent
}

// ---------------------------------------------------------------------------
extern "C" void kernel_launch(void* const* d_in, const int* in_sizes, int n_in,
                              void* d_out, int out_size, void* d_ws,
                              size_t ws_size, hipStream_t stream) {
  const float* d1   = (const float*)d_in[0];
  const float* d2   = (const float*)d_in[1];
  const float* invT = (const float*)d_in[2];
  float* out = (float*)d_out;

  char* wsp = (char*)d_ws;
  auto alloc = [&](size_t bytes) -> void* {
    void* p = (void*)wsp;
    wsp += (bytes + 255) & ~(size_t)255;
    return p;
  };
  float* sq1 = (float*)alloc((size_t)N_ * 4);
  float* sq2 = (float*)alloc((size_t)M_ * 4);
  float* rmp = (float*)alloc((size_t)CT * N_ * 4);
  float* rsp = (float*)alloc((size_t)CT * N_ * 4);
  int*   rap = (int*)  alloc((size_t)CT * N_ * 4);
  float* cmp_ = (float*)alloc((size_t)RT * M_ * 4);
  float* csp  = (float*)alloc((size_t)RT * M_ * 4);
  int*   cap  = (int*)  alloc((size_t)RT * M_ * 4);
  float* rowOff = (float*)alloc((size_t)N_ * 4);
  int*   rowArg = (int*)  alloc((size_t)N_ * 4);
  float* colOff = (float*)alloc((size_t)M_ * 4);
  int*   colArg = (int*)  alloc((size_t)M_ * 4);

  sqnorm_kernel<<<(N_ + M_ + 255) / 256, 256, 0, stream>>>(d1, d2, sq1, sq2);

  dim3 grid(CT, RT);
  gemm_stats_kernel<<<grid, 256, 0, stream>>>(d1, d2, sq1, sq2, invT, rmp, rsp,
                                              rap, cmp_, csp, cap);
  reduce_stats_kernel<<<(N_ + 255) / 256, 256, 0, stream>>>(rmp, rsp, rap, N_,
                                                            CT, rowOff, rowArg);
  reduce_stats_kernel<<<(M_ + 255) / 256, 256, 0, stream>>>(cmp_, csp, cap, M_,
                                                            RT, colOff, colArg);
  gemm_write_kernel<<<grid, 256, 0, stream>>>(d1, d2, sq1, sq2, invT, rowOff,
                                              colOff, out,
                                              out + (size_t)N_ * M_);
  matches_kernel<<<(N_ + 255) / 256, 256, 0, stream>>>(
      rowArg, colArg, out + 2ull * N_ * M_);
}